// QuantumCircuitLayer_15375982920210
// MI455X (gfx1250) — compile-verified
//
#include <hip/hip_runtime.h>
#include <math.h>

typedef __attribute__((ext_vector_type(2))) float v2f;
typedef __attribute__((ext_vector_type(8))) float v8f;

#define NQ 8
#define NL 4
#define NO 16

// Branch-free tanh: prefer the gfx1250 hardware V_TANH_F32 if clang declares
// the builtin; otherwise tanh(x) = 1 - 2/(exp(2x)+1), which is valid over the
// full range (exp->0 => -1, exp->inf => rcp gives 0 => +1) with no branches
// and therefore no EXEC save/restore around the WMMAs.
__device__ __forceinline__ float fast_tanh(float x) {
#if __has_builtin(__builtin_amdgcn_tanhf)
    return __builtin_amdgcn_tanhf(x);
#else
    const float e = __expf(2.0f * x);                       // native v_exp_f32
    return fmaf(-2.0f, __builtin_amdgcn_rcpf(e + 1.0f), 1.0f);
#endif
}

// ---------------------------------------------------------------------------
// Setup: fold 4 circuit layers + projection into W_eff^T (8x16) in d_ws.
// The entire post-tanh circuit is linear, so W_eff = proj_w (16x8) @ M (8x8),
// M = product of per-layer diag(scale) and bidiagonal entangle mixes.
// ---------------------------------------------------------------------------
__global__ void qc_setup(const float* __restrict__ rot,   // [4][8][3]
                         const float* __restrict__ ent,   // [4][7]
                         const float* __restrict__ pw,    // [16][8]
                         float* __restrict__ wt)          // out: [8][16] = W_eff^T
{
    const int o = threadIdx.x;
    if (o >= NO) return;

    float scale[NL][NQ];
    float w[NL][NQ - 1];
    for (int l = 0; l < NL; ++l) {
        for (int q = 0; q < NQ; ++q) {
            const float* r = rot + (l * NQ + q) * 3;
            float s = 1.0f;
            for (int j = 0; j < 3; ++j) s *= (cosf(r[j]) + sinf(r[j]));
            scale[l][q] = s;
        }
        for (int i = 0; i < NQ - 1; ++i) {
            const float e = ent[l * (NQ - 1) + i];
            w[l][i] = 1.0f / (1.0f + expf(-e));
        }
    }

    // M[row][col]: image of basis vector e_col through all layers
    float M[NQ][NQ];
    for (int c = 0; c < NQ; ++c) {
        float v[NQ];
        for (int q = 0; q < NQ; ++q) v[q] = (q == c) ? 1.0f : 0.0f;
        for (int l = 0; l < NL; ++l) {
            for (int q = 0; q < NQ; ++q) v[q] *= scale[l][q];
            float nv[NQ];
            for (int i = 0; i < NQ - 1; ++i)
                nv[i] = v[i] * (1.0f - w[l][i]) + v[i + 1] * w[l][i];
            nv[NQ - 1] = v[NQ - 1];
            for (int q = 0; q < NQ; ++q) v[q] = nv[q];
        }
        for (int q = 0; q < NQ; ++q) M[q][c] = v[q];
    }

    for (int q = 0; q < NQ; ++q) {
        float acc = 0.0f;
        for (int k = 0; k < NQ; ++k) acc = fmaf(pw[o * NQ + k], M[k][q], acc);
        wt[q * NO + o] = acc;   // W_eff^T[q][o]
    }
}

// ---------------------------------------------------------------------------
// Main: per wave, 16 sites/tile.  D(16x16) = tanh(A 16x8) x W_eff^T(8x16) + b
// via two chained V_WMMA_F32_16X16X4_F32 (full f32 precision).
//
// A layout (16x4 f32 WMMA): lane L holds row M = L%16; lanes 0-15 carry
// internal K0,K1; lanes 16-31 carry K2,K3.  Logical columns {0,1,4,5} feed
// WMMA#1 and {2,3,6,7} feed WMMA#2 so each lane needs exactly one contiguous
// float4 load: x[row*64 + (L/16)*4 .. +3]  -> only the 8 needed floats of the
// 64-wide feature row are ever read (16 MiB instead of 128 MiB).
// B = W_eff^T loaded once per wave with the matching K-permutation.
// C is pre-filled with proj_b[n], so bias rides the accumulator for free.
// ---------------------------------------------------------------------------
__global__ void __launch_bounds__(256) qc_main(const float* __restrict__ x,
                                               const float* __restrict__ wt,
                                               const float* __restrict__ pb,
                                               float* __restrict__ out,
                                               int ntiles)
{
    const int lane = threadIdx.x & 31;
    const int n    = lane & 15;   // output column / B column
    const int g    = lane >> 4;   // half-wave group (internal K pair select)

    const int wave   = (int)((blockIdx.x * blockDim.x + threadIdx.x) >> 5);
    const int nwaves = (int)((gridDim.x * blockDim.x) >> 5);

    // B operands: internal rows (K0,K1 | K2,K3) = logical rows
    //   WMMA#1: (0,1 | 4,5)   WMMA#2: (2,3 | 6,7)
    v2f b1, b2;
    b1.x = wt[(4 * g + 0) * NO + n];
    b1.y = wt[(4 * g + 1) * NO + n];
    b2.x = wt[(4 * g + 2) * NO + n];
    b2.y = wt[(4 * g + 3) * NO + n];

    const float bias = pb[n];
    v8f cbias;
#pragma unroll
    for (int r = 0; r < 8; ++r) cbias[r] = bias;

    for (int t = wave; t < ntiles; t += nwaves) {
        const long rowBase = (long)t * 16;
        const long row     = rowBase + n;

        const float4 xv = *(const float4*)(x + row * 64 + g * 4);

        // prefetch next tile's data for this lane (gfx1250 global_prefetch_b8)
        __builtin_prefetch(x + (row + (long)nwaves * 16) * 64 + g * 4, 0, 1);

        v2f a1, a2;
        a1.x = fast_tanh(xv.x);
        a1.y = fast_tanh(xv.y);
        a2.x = fast_tanh(xv.z);
        a2.y = fast_tanh(xv.w);

        v8f acc = __builtin_amdgcn_wmma_f32_16x16x4_f32(
            false, a1, false, b1, (short)0, cbias, false, false);
        acc = __builtin_amdgcn_wmma_f32_16x16x4_f32(
            false, a2, false, b2, (short)0, acc, false, false);

        // D: lane n = column, reg r -> row (r + 8*g); store 16 outputs/row
        float* po = out + (rowBase + (long)(8 * g)) * 16 + n;
#pragma unroll
        for (int r = 0; r < 8; ++r) po[r * 16] = acc[r];
    }
}

// ---------------------------------------------------------------------------
extern "C" void kernel_launch(void* const* d_in, const int* in_sizes, int n_in,
                              void* d_out, int out_size, void* d_ws, size_t ws_size,
                              hipStream_t stream) {
    const float* x   = (const float*)d_in[0];  // [64,8192,64]
    const float* rot = (const float*)d_in[1];  // [4,8,3]
    const float* ent = (const float*)d_in[2];  // [4,7]
    const float* pw  = (const float*)d_in[3];  // [16,8]
    const float* pb  = (const float*)d_in[4];  // [16]
    float*       out = (float*)d_out;          // [64,8192,16]
    float*       wt  = (float*)d_ws;           // W_eff^T: 8*16 floats

    qc_setup<<<1, 32, 0, stream>>>(rot, ent, pw, wt);

    const long rows   = (long)in_sizes[0] / 64;   // 524288 sites
    const int  ntiles = (int)(rows / 16);         // 32768 16-row tiles

    qc_main<<<512, 256, 0, stream>>>(x, wt, pb, out, ntiles);
}